// SelectiveScanPlugin_64922725646540
// MI455X (gfx1250) — compile-verified
//
#include <hip/hip_runtime.h>
#include <math.h>

#define BB      32
#define TT      8192
#define DM      16
#define KG      16
#define HID     64
#define NTOK    60
#define NCHUNK  128
#define CLEN    (TT / NCHUNK)   // 64 steps per chunk

typedef float v2f __attribute__((ext_vector_type(2)));
typedef float v8f __attribute__((ext_vector_type(8)));

// ---------------------------------------------------------------------------
// One 16x16x16 f32 matmul  S_new = M @ S  using 4 chained V_WMMA_F32_16X16X4_F32.
// a8[j*2+v] holds M[lane&15][4j + v + 2*(lane>>4)]   (A-operand layout, ISA 7.12.2)
// S is in C/D layout: S[v] = elem(r = v + 8*(lane>>4), c = lane&15).
// B-slice j needs rows 4j..4j+3 of S -> gathered cross-lane with ds_bpermute.
// ---------------------------------------------------------------------------
__device__ __forceinline__ v8f mat_chain_mul(v8f S, const float* a8,
                                             int lane, int cn, int hi) {
  v8f c = {};
  #pragma unroll
  for (int j = 0; j < 4; ++j) {
    v2f a;
    a[0] = a8[j * 2 + 0];
    a[1] = a8[j * 2 + 1];
    v2f b;
    #pragma unroll
    for (int v = 0; v < 2; ++v) {
      const int rlo = 4 * j + v;      // row needed by lanes 0..15
      const int rhi = rlo + 2;        // row needed by lanes 16..31
      // row r of S lives in register S[r&7] on half-wave (r>>3)
      float blo = __shfl(S[rlo & 7], ((rlo >> 3) << 4) + cn, 32);
      float bhi = __shfl(S[rhi & 7], ((rhi >> 3) << 4) + cn, 32);
      b[v] = (lane < 16) ? blo : bhi;
    }
    // (neg_a, A, neg_b, B, c_mod, C, reuse_a, reuse_b)
    c = __builtin_amdgcn_wmma_f32_16x16x4_f32(false, a, false, b,
                                              (short)0, c, false, false);
  }
  return c;
}

// ---------------------------------------------------------------------------
// Phase 1: mixture weights  w[b,t,k] = softmax_k( emb[id] . Wk[:,k] + bk[k] )
// Pure elementwise/bandwidth kernel: one thread per (b,t).
// ---------------------------------------------------------------------------
__global__ void mix_weights_kernel(const int* __restrict__ ids,
                                   const float* __restrict__ emb,
                                   const float* __restrict__ Wk,
                                   const float* __restrict__ bk,
                                   float* __restrict__ w) {
  int i = blockIdx.x * blockDim.x + threadIdx.x;
  if (i >= BB * TT) return;
  const int id = ids[i];
  const float* x = emb + (size_t)id * DM;
  float lg[KG];
  #pragma unroll
  for (int k = 0; k < KG; ++k) lg[k] = bk[k];
  #pragma unroll
  for (int d = 0; d < DM; ++d) {
    const float xd = x[d];
    #pragma unroll
    for (int k = 0; k < KG; ++k) lg[k] = fmaf(xd, Wk[d * KG + k], lg[k]);
  }
  float mx = lg[0];
  #pragma unroll
  for (int k = 1; k < KG; ++k) mx = fmaxf(mx, lg[k]);
  float s = 0.f;
  #pragma unroll
  for (int k = 0; k < KG; ++k) { lg[k] = __expf(lg[k] - mx); s += lg[k]; }
  const float inv = 1.0f / s;
  #pragma unroll
  for (int k = 0; k < KG; ++k) w[(size_t)i * KG + k] = lg[k] * inv;
}

// ---------------------------------------------------------------------------
// Phase 2: per-chunk partial products. One wave per chunk (4096 waves).
// P_c = M_{t0+CLEN-1} @ ... @ M_{t0}, built with WMMA f32 16x16x4.
// Generators live in 128 VGPRs per lane, pre-layouted for the A operand.
// ---------------------------------------------------------------------------
__global__ void __launch_bounds__(32)
chunk_products_kernel(const float* __restrict__ w,
                      const float* __restrict__ G,
                      float* __restrict__ P) {
  const int chunk = blockIdx.x;                 // b * NCHUNK + c
  const int b  = chunk / NCHUNK;
  const int c  = chunk % NCHUNK;
  const int lane = threadIdx.x;
  const int cn = lane & 15;
  const int hi = lane >> 4;

  // Preload all 16 generators, each lane keeps the 8 elements it will need
  // in A-operand layout: G[k][cn][4j + v + 2*hi]
  float g[KG][8];
  #pragma unroll
  for (int k = 0; k < KG; ++k)
    #pragma unroll
    for (int j = 0; j < 4; ++j)
      #pragma unroll
      for (int v = 0; v < 2; ++v)
        g[k][j * 2 + v] = G[k * 256 + cn * 16 + 4 * j + v + 2 * hi];

  // S = identity in C/D layout
  v8f S = {};
  #pragma unroll
  for (int v = 0; v < 8; ++v) S[v] = ((v + 8 * hi) == cn) ? 1.0f : 0.0f;

  // wave-uniform pointer -> scalar loads of the 16 mixture weights per step
  const float* wp = w + ((size_t)b * TT + (size_t)c * CLEN) * KG;

  for (int t = 0; t < CLEN; ++t) {
    __builtin_prefetch(wp + 4 * KG, 0, 0);      // global_prefetch next steps' weights
    // Build M_t directly in A layout: M = sum_k w_k * G_k  (w_k wave-uniform)
    float a8[8];
    #pragma unroll
    for (int s = 0; s < 8; ++s) {
      float m = 0.f;
      #pragma unroll
      for (int k = 0; k < KG; ++k) m = fmaf(wp[k], g[k][s], m);
      a8[s] = m;
    }
    S = mat_chain_mul(S, a8, lane, cn, hi);     // S = M_t @ S   (4x v_wmma)
    wp += KG;
  }

  // store chunk product row-major
  float* Pp = P + (size_t)chunk * 256;
  #pragma unroll
  for (int v = 0; v < 8; ++v) Pp[(v + 8 * hi) * 16 + cn] = S[v];
}

// ---------------------------------------------------------------------------
// Phase 3: sequential combine of the 128 chunk products per batch (WMMA),
// then fused MLP readout 256 -> 64 -> 60 through LDS. One wave per batch.
// ---------------------------------------------------------------------------
__global__ void __launch_bounds__(32)
combine_readout_kernel(const float* __restrict__ P,
                       const float* __restrict__ W1,
                       const float* __restrict__ b1,
                       const float* __restrict__ W2,
                       const float* __restrict__ b2,
                       float* __restrict__ out) {
  const int b = blockIdx.x;
  const int lane = threadIdx.x;
  const int cn = lane & 15;
  const int hi = lane >> 4;
  const float* Pb = P + (size_t)b * NCHUNK * 256;

  // S = P_0 into C/D layout
  v8f S;
  #pragma unroll
  for (int v = 0; v < 8; ++v) S[v] = Pb[(v + 8 * hi) * 16 + cn];

  for (int ci = 1; ci < NCHUNK; ++ci) {
    const float* Pc = Pb + (size_t)ci * 256;
    float a8[8];                                // load P_ci in A layout
    #pragma unroll
    for (int j = 0; j < 4; ++j)
      #pragma unroll
      for (int v = 0; v < 2; ++v)
        a8[j * 2 + v] = Pc[cn * 16 + 4 * j + v + 2 * hi];
    S = mat_chain_mul(S, a8, lane, cn, hi);     // S = P_ci @ S
  }

  // ---- fused readout MLP ----
  __shared__ float sh[256 + HID];
  #pragma unroll
  for (int v = 0; v < 8; ++v) sh[(v + 8 * hi) * 16 + cn] = S[v];
  __syncthreads();

  #pragma unroll
  for (int r = 0; r < 2; ++r) {                 // 64 hidden units / 32 lanes
    const int jj = lane + 32 * r;
    float h = b1[jj];
    for (int i = 0; i < 256; ++i) h = fmaf(sh[i], W1[i * HID + jj], h);
    sh[256 + jj] = fmaxf(h, 0.0f);              // relu
  }
  __syncthreads();

  #pragma unroll
  for (int r = 0; r < 2; ++r) {
    const int n = lane + 32 * r;
    if (n < NTOK) {
      float acc = b2[n];
      for (int j = 0; j < HID; ++j) acc = fmaf(sh[256 + j], W2[j * NTOK + n], acc);
      out[b * NTOK + n] = acc;
    }
  }
}

// ---------------------------------------------------------------------------
extern "C" void kernel_launch(void* const* d_in, const int* in_sizes, int n_in,
                              void* d_out, int out_size, void* d_ws, size_t ws_size,
                              hipStream_t stream) {
  const int*   ids = (const int*)  d_in[0];
  const float* emb = (const float*)d_in[1];
  const float* Wk  = (const float*)d_in[2];
  const float* bk  = (const float*)d_in[3];
  const float* G   = (const float*)d_in[4];
  const float* W1  = (const float*)d_in[5];
  const float* b1  = (const float*)d_in[6];
  const float* W2  = (const float*)d_in[7];
  const float* b2  = (const float*)d_in[8];
  float* out = (float*)d_out;

  // workspace: w [B*T*16] floats (16 MB) then P [B*NCHUNK*256] floats (4 MB)
  float* w = (float*)d_ws;
  float* P = w + (size_t)BB * TT * KG;

  mix_weights_kernel<<<(BB * TT + 255) / 256, 256, 0, stream>>>(ids, emb, Wk, bk, w);
  chunk_products_kernel<<<BB * NCHUNK, 32, 0, stream>>>(w, G, P);
  combine_readout_kernel<<<BB, 32, 0, stream>>>(P, W1, b1, W2, b2, out);
}